// GatedGCN_57775900066610
// MI455X (gfx1250) — compile-verified
//
#include <hip/hip_runtime.h>

#define N_NODES 8192
#define N_EDGES 524288
#define DIM 64
#define LAYERS 9
#define MLP_DEPTH 3

typedef __attribute__((ext_vector_type(2))) float v2f;
typedef __attribute__((ext_vector_type(8))) float v8f;

// h0 = pad(node_feat [N,1] -> [N,64])
__global__ void k_init_h(const float* __restrict__ nf, float* __restrict__ h) {
    int i = blockIdx.x * blockDim.x + threadIdx.x;
    int n = i >> 6, d = i & 63;
    h[i] = (d == 0) ? nf[n] : 0.0f;
}

__global__ void k_zero(float* __restrict__ p) {
    int i = blockIdx.x * blockDim.x + threadIdx.x;
    p[i] = 0.0f;
}

// C[M x Ncols] = act(A[M x 64] @ W[Ncols x 64]^T + bias), M = 8192.
// One wave per 16x16 output tile. fp32 WMMA, K split into 16 steps of 4.
__global__ void k_gemm_wmma(const float* __restrict__ A,
                            const float* __restrict__ W,
                            const float* __restrict__ bias,
                            float* __restrict__ C,
                            int tilesN, int relu) {
    int wave = (blockIdx.x * blockDim.x + threadIdx.x) >> 5;
    int lane = threadIdx.x & 31;
    int tm = wave / tilesN;
    int tn = wave % tilesN;
    int half = lane >> 4;
    int l    = lane & 15;
    const float* arow = A + (size_t)(tm * 16 + l) * DIM;   // A row m0+l
    const float* wrow = W + (size_t)(tn * 16 + l) * DIM;   // W row n0+l (B is W^T)
    v8f acc = {};
#pragma unroll
    for (int s = 0; s < 16; ++s) {
        int off = 4 * s + 2 * half;
        v2f av, bv;
        av.x = arow[off]; av.y = arow[off + 1];
        bv.x = wrow[off]; bv.y = wrow[off + 1];
        acc = __builtin_amdgcn_wmma_f32_16x16x4_f32(
            false, av, false, bv, (short)0, acc, false, false);
    }
    float bb = bias[tn * 16 + l];
    int ldc = tilesN * 16;
#pragma unroll
    for (int j = 0; j < 8; ++j) {
        float v = acc[j] + bb;
        if (relu) v = fmaxf(v, 0.0f);
        C[(size_t)(tm * 16 + j + 8 * half) * ldc + tn * 16 + l] = v;
    }
}

// a[dst] += hl[src] over E real edges + N self loops. One wave per edge,
// each lane handles 2 floats. HW fp32 atomics forced via inline CDNA5 asm:
// global_atomic_add_f32 (no-return form, STOREcnt-tracked, device scope).
__global__ void k_scatter(const float* __restrict__ hl,
                          const int* __restrict__ src,
                          const int* __restrict__ dst,
                          float* __restrict__ a, int nreal, int ntot) {
    int gid = blockIdx.x * blockDim.x + threadIdx.x;
    int e    = gid >> 5;
    int lane = threadIdx.x & 31;
    if (e >= ntot) return;
    int s_, d_;
    if (e < nreal) { s_ = src[e]; d_ = dst[e]; }
    else           { s_ = e - nreal; d_ = s_; }
    const float2* sp = (const float2*)(hl + (size_t)s_ * DIM);
    float2 v = sp[lane];
    float* dp = a + (size_t)d_ * DIM + lane * 2;
    asm volatile(
        "global_atomic_add_f32 %0, %1, off scope:SCOPE_DEV\n\t"
        "global_atomic_add_f32 %0, %2, off offset:4 scope:SCOPE_DEV"
        :
        : "v"(dp), "v"(v.x), "v"(v.y)
        : "memory");
}

// GRUCell gates (r,z,n order) + relu, h updated in place.
__global__ void k_gru(const float* __restrict__ gi,
                      const float* __restrict__ gh,
                      float* __restrict__ h) {
    int i = blockIdx.x * blockDim.x + threadIdx.x;   // over N*64
    int n = i >> 6, d = i & 63;
    const float* gin = gi + (size_t)n * 3 * DIM;
    const float* ghn = gh + (size_t)n * 3 * DIM;
    float i_r = gin[d], i_z = gin[DIM + d], i_n = gin[2 * DIM + d];
    float h_r = ghn[d], h_z = ghn[DIM + d], h_n = ghn[2 * DIM + d];
    float r  = 1.0f / (1.0f + expf(-(i_r + h_r)));
    float z  = 1.0f / (1.0f + expf(-(i_z + h_z)));
    float nn = tanhf(i_n + r * h_n);
    float hv = h[i];
    float hnew = (1.0f - z) * nn + z * hv;
    h[i] = fmaxf(hnew, 0.0f);
}

// out[8192,8192] = H @ H^T. One wave per 16x64 strip (A frag reused x4).
__global__ void k_outer_wmma(const float* __restrict__ H, float* __restrict__ C) {
    int wave = (blockIdx.x * blockDim.x + threadIdx.x) >> 5;
    int lane = threadIdx.x & 31;
    int tm = wave >> 7;       // 512 row tiles of 16
    int tg = wave & 127;      // 128 col groups of 64
    int half = lane >> 4, l = lane & 15;
    const float* arow = H + (size_t)(tm * 16 + l) * DIM;
    const float* b0 = H + (size_t)(tg * 64 +  0 + l) * DIM;
    const float* b1 = H + (size_t)(tg * 64 + 16 + l) * DIM;
    const float* b2 = H + (size_t)(tg * 64 + 32 + l) * DIM;
    const float* b3 = H + (size_t)(tg * 64 + 48 + l) * DIM;
    v8f c0 = {}, c1 = {}, c2 = {}, c3 = {};
#pragma unroll
    for (int s = 0; s < 16; ++s) {
        int off = 4 * s + 2 * half;
        v2f av, v0, v1, v2, v3;
        av.x = arow[off]; av.y = arow[off + 1];
        v0.x = b0[off];   v0.y = b0[off + 1];
        v1.x = b1[off];   v1.y = b1[off + 1];
        v2.x = b2[off];   v2.y = b2[off + 1];
        v3.x = b3[off];   v3.y = b3[off + 1];
        c0 = __builtin_amdgcn_wmma_f32_16x16x4_f32(false, av, false, v0, (short)0, c0, false, false);
        c1 = __builtin_amdgcn_wmma_f32_16x16x4_f32(false, av, false, v1, (short)0, c1, false, false);
        c2 = __builtin_amdgcn_wmma_f32_16x16x4_f32(false, av, false, v2, (short)0, c2, false, false);
        c3 = __builtin_amdgcn_wmma_f32_16x16x4_f32(false, av, false, v3, (short)0, c3, false, false);
    }
    size_t base = (size_t)(tm * 16) * N_NODES + (size_t)tg * 64;
#pragma unroll
    for (int j = 0; j < 8; ++j) {
        size_t r = base + (size_t)(j + 8 * half) * N_NODES + l;
        C[r]      = c0[j];
        C[r + 16] = c1[j];
        C[r + 32] = c2[j];
        C[r + 48] = c3[j];
    }
}

extern "C" void kernel_launch(void* const* d_in, const int* in_sizes, int n_in,
                              void* d_out, int out_size, void* d_ws, size_t ws_size,
                              hipStream_t stream) {
    (void)in_sizes; (void)n_in; (void)out_size; (void)ws_size;
    const float* node_feat = (const float*)d_in[0];
    const int*   edge_src  = (const int*)d_in[1];
    const int*   edge_dst  = (const int*)d_in[2];
    const float* conv_W    = (const float*)d_in[3];
    const float* conv_b    = (const float*)d_in[4];
    const float* gru_Wih   = (const float*)d_in[5];
    const float* gru_Whh   = (const float*)d_in[6];
    const float* gru_bih   = (const float*)d_in[7];
    const float* gru_bhh   = (const float*)d_in[8];
    const float* mlp_W     = (const float*)d_in[9];
    const float* mlp_b     = (const float*)d_in[10];
    float* out = (float*)d_out;

    // workspace layout (floats): h | hl | a | gi | gh  = 18 MB total
    float* h  = (float*)d_ws;
    float* hl = h  + (size_t)N_NODES * DIM;
    float* a  = hl + (size_t)N_NODES * DIM;
    float* gi = a  + (size_t)N_NODES * DIM;
    float* gh = gi + (size_t)N_NODES * 3 * DIM;

    dim3 blk(256);
    const int elemBlocks = (N_NODES * DIM) / 256;      // 2048
    const int ntot = N_EDGES + N_NODES;                // 532480 (multiple of 8)

    k_init_h<<<elemBlocks, blk, 0, stream>>>(node_feat, h);

    for (int lyr = 0; lyr < LAYERS; ++lyr) {
        // hl = h @ W^T + b   (512x4 tiles, 8 waves/block)
        k_gemm_wmma<<<(512 * 4) / 8, blk, 0, stream>>>(
            h, conv_W + (size_t)lyr * DIM * DIM, conv_b + (size_t)lyr * DIM, hl, 4, 0);
        // a = segment_sum(hl[src], dst)
        k_zero<<<elemBlocks, blk, 0, stream>>>(a);
        k_scatter<<<ntot / 8, blk, 0, stream>>>(hl, edge_src, edge_dst, a, N_EDGES, ntot);
        // gi = a @ Wih^T + bih ; gh = h @ Whh^T + bhh   (512x12 tiles each)
        k_gemm_wmma<<<(512 * 12) / 8, blk, 0, stream>>>(
            a, gru_Wih + (size_t)lyr * 3 * DIM * DIM, gru_bih + (size_t)lyr * 3 * DIM, gi, 12, 0);
        k_gemm_wmma<<<(512 * 12) / 8, blk, 0, stream>>>(
            h, gru_Whh + (size_t)lyr * 3 * DIM * DIM, gru_bhh + (size_t)lyr * 3 * DIM, gh, 12, 0);
        // h = relu(GRU(a-derived gates, h)) in place
        k_gru<<<elemBlocks, blk, 0, stream>>>(gi, gh, h);
    }

    // MLP: (Linear -> ReLU) x3, ping-pong h <-> hl
    float* cur = h; float* oth = hl;
    for (int i = 0; i < MLP_DEPTH; ++i) {
        k_gemm_wmma<<<(512 * 4) / 8, blk, 0, stream>>>(
            cur, mlp_W + (size_t)i * DIM * DIM, mlp_b + (size_t)i * DIM, oth, 4, 1);
        float* t = cur; cur = oth; oth = t;
    }

    // out = cur @ cur^T  (512 x 128 waves)
    k_outer_wmma<<<(512 * 128) / 8, blk, 0, stream>>>(cur, out);
}